// SSMBlock_13477607375793
// MI455X (gfx1250) — compile-verified
//
#include <hip/hip_runtime.h>
#include <math.h>

// ---- problem constants (from reference) ----
#define D_MODEL  1024
#define D_STATE  16
#define D_CONV   4
#define D_INNER  2048          // EXPAND * D_MODEL
#define DT_RANK  64            // (1024+15)//16
#define NBATCH   2
#define SEQ      2048
#define MROWS    (NBATCH * SEQ)   // 4096 token rows
#define XPROJ_N  (DT_RANK + 2 * D_STATE)  // 96

typedef float v2f __attribute__((ext_vector_type(2)));
typedef float v8f __attribute__((ext_vector_type(8)));
typedef int   v4i __attribute__((ext_vector_type(4)));

// 32 cols + pad: keeps 16B alignment of every (row,4k-col) slot and is
// bank-conflict free (36*m mod 64 distinct for m=0..15)
#define LDS_STRIDE 36

__device__ __forceinline__ float silu_f(float v) {
    return v * (1.0f / (1.0f + __expf(-v)));
}
__device__ __forceinline__ float softplus_f(float v) {
    return (v > 20.0f) ? v : log1pf(__expf(v));
}

// ---------------- CDNA5 async global->LDS path (guarded) ----------------
#if __has_builtin(__builtin_amdgcn_global_load_async_to_lds_b128)
#define HAVE_ASYNC_LDS 1
typedef __attribute__((address_space(1))) v4i gv4i_t;   // global v4i
typedef __attribute__((address_space(3))) v4i lv4i_t;   // LDS v4i
// Cast through integers: generic->AS1 is bit-identical for global memory;
// generic LDS address low 32 bits are the LDS offset (AS3 pointers are 32-bit).
#define ASYNC_B128(ldsp, gp)                                                  \
    __builtin_amdgcn_global_load_async_to_lds_b128(                           \
        (gv4i_t*)(unsigned long long)(const void*)(gp),                       \
        (lv4i_t*)(unsigned int)(unsigned long long)(void*)(ldsp), 0, 0)
#if __has_builtin(__builtin_amdgcn_s_wait_asynccnt)
#define WAIT_ASYNC(n) __builtin_amdgcn_s_wait_asynccnt(n)
#else
#define WAIT_ASYNC(n) asm volatile("s_wait_asynccnt %0" ::"i"(n) : "memory")
#endif
#else
#define HAVE_ASYNC_LDS 0
#endif

// =====================================================================
// Generic fp32 WMMA GEMM:  C[M x N] = A[M x K] (row-major, stride lda)
//                                    * W[N x K]^T (row-major, stride K)
// M = MROWS (multiple of 128), N guarded, K multiple of 32.
// epi: 0 = plain store, 1 = softplus(acc + bias[n])
// Block: 256 threads (8 waves), block tile 128x64, wave tile 32x32
// (2x2 grid of 16x16 WMMA accumulators => 4 wmma per 4 LDS frag loads).
// Double-buffered LDS fed by GLOBAL_LOAD_ASYNC_TO_LDS_B128 when available.
// =====================================================================
__global__ __launch_bounds__(256)
void gemm_wmma_f32(const float* __restrict__ A, int lda,
                   const float* __restrict__ W,
                   const float* __restrict__ bias,
                   float* __restrict__ C,
                   int N, int K, int epi)
{
    __shared__ float Al[2][128 * LDS_STRIDE];   // 36 KB
    __shared__ float Bl[2][64 * LDS_STRIDE];    // 18 KB

    const int tid  = threadIdx.x;
    const int lane = tid & 31;
    const int wv   = tid >> 5;
    const int m    = lane & 15;     // row(A)/col(B) within a 16-tile
    const int kh   = lane >> 4;     // K-half: lanes 16-31 hold K=2,3
    const int wr   = wv & 3;        // wave row   (0..3) -> 32 rows each
    const int wc   = wv >> 2;       // wave col   (0..1) -> 32 cols each

    const int m0 = blockIdx.y * 128;
    const int n0 = blockIdx.x * 64;

    v8f acc00 = {}, acc01 = {}, acc10 = {}, acc11 = {};

#if HAVE_ASYNC_LDS
    // ---- issue chunk 0 ----
    {
        #pragma unroll
        for (int it = 0; it < 4; ++it) {           // A: 128x32 = 1024 float4
            const int i = tid + it * 256;
            const int row = i >> 3, c4 = (i & 7) << 2;
            ASYNC_B128(&Al[0][row * LDS_STRIDE + c4],
                       A + (size_t)(m0 + row) * (size_t)lda + c4);
        }
        #pragma unroll
        for (int it = 0; it < 2; ++it) {           // B: 64x32 = 512 float4
            const int i = tid + it * 256;
            const int row = i >> 3, c4 = (i & 7) << 2;
            float* pb = &Bl[0][row * LDS_STRIDE + c4];
            if (n0 + row < N) {
                ASYNC_B128(pb, W + (size_t)(n0 + row) * (size_t)K + c4);
            } else {
                pb[0] = 0.f; pb[1] = 0.f; pb[2] = 0.f; pb[3] = 0.f;
            }
        }
    }
    int buf = 0;
    for (int k0 = 0; k0 < K; k0 += 32, buf ^= 1) {
        if (k0 + 32 < K) {
            const int nb = buf ^ 1, kn = k0 + 32;
            #pragma unroll
            for (int it = 0; it < 4; ++it) {
                const int i = tid + it * 256;
                const int row = i >> 3, c4 = (i & 7) << 2;
                ASYNC_B128(&Al[nb][row * LDS_STRIDE + c4],
                           A + (size_t)(m0 + row) * (size_t)lda + kn + c4);
            }
            #pragma unroll
            for (int it = 0; it < 2; ++it) {
                const int i = tid + it * 256;
                const int row = i >> 3, c4 = (i & 7) << 2;
                float* pb = &Bl[nb][row * LDS_STRIDE + c4];
                if (n0 + row < N) {
                    ASYNC_B128(pb, W + (size_t)(n0 + row) * (size_t)K + kn + c4);
                } else {
                    pb[0] = 0.f; pb[1] = 0.f; pb[2] = 0.f; pb[3] = 0.f;
                }
            }
            WAIT_ASYNC(6);   // 6 async ops just issued; current chunk complete
        } else {
            WAIT_ASYNC(0);
        }
        __syncthreads();

        const float* __restrict__ Ac = Al[buf];
        const float* __restrict__ Bc = Bl[buf];
        #pragma unroll
        for (int kk = 0; kk < 32; kk += 4) {
            const int ko = kk + 2 * kh;
            v2f a0 = *(const v2f*)&Ac[(wr * 32 + m)      * LDS_STRIDE + ko];
            v2f a1 = *(const v2f*)&Ac[(wr * 32 + 16 + m) * LDS_STRIDE + ko];
            v2f b0 = *(const v2f*)&Bc[(wc * 32 + m)      * LDS_STRIDE + ko];
            v2f b1 = *(const v2f*)&Bc[(wc * 32 + 16 + m) * LDS_STRIDE + ko];
            acc00 = __builtin_amdgcn_wmma_f32_16x16x4_f32(false, a0, false, b0, (short)0, acc00, false, false);
            acc01 = __builtin_amdgcn_wmma_f32_16x16x4_f32(false, a0, false, b1, (short)0, acc01, false, false);
            acc10 = __builtin_amdgcn_wmma_f32_16x16x4_f32(false, a1, false, b0, (short)0, acc10, false, false);
            acc11 = __builtin_amdgcn_wmma_f32_16x16x4_f32(false, a1, false, b1, (short)0, acc11, false, false);
        }
        __syncthreads();
    }
#else
    // ---- fallback: synchronous VGPR-staged tile copies, single buffer ----
    for (int k0 = 0; k0 < K; k0 += 32) {
        #pragma unroll
        for (int it = 0; it < 4; ++it) {
            const int i = tid + it * 256;
            const int row = i >> 3, c4 = (i & 7) << 2;
            const float4 va = *(const float4*)(A + (size_t)(m0 + row) * (size_t)lda + k0 + c4);
            float* pa = &Al[0][row * LDS_STRIDE + c4];
            pa[0] = va.x; pa[1] = va.y; pa[2] = va.z; pa[3] = va.w;
        }
        #pragma unroll
        for (int it = 0; it < 2; ++it) {
            const int i = tid + it * 256;
            const int row = i >> 3, c4 = (i & 7) << 2;
            float4 vb = make_float4(0.f, 0.f, 0.f, 0.f);
            if (n0 + row < N)
                vb = *(const float4*)(W + (size_t)(n0 + row) * (size_t)K + k0 + c4);
            float* pb = &Bl[0][row * LDS_STRIDE + c4];
            pb[0] = vb.x; pb[1] = vb.y; pb[2] = vb.z; pb[3] = vb.w;
        }
        __syncthreads();
        #pragma unroll
        for (int kk = 0; kk < 32; kk += 4) {
            const int ko = kk + 2 * kh;
            v2f a0 = *(const v2f*)&Al[0][(wr * 32 + m)      * LDS_STRIDE + ko];
            v2f a1 = *(const v2f*)&Al[0][(wr * 32 + 16 + m) * LDS_STRIDE + ko];
            v2f b0 = *(const v2f*)&Bl[0][(wc * 32 + m)      * LDS_STRIDE + ko];
            v2f b1 = *(const v2f*)&Bl[0][(wc * 32 + 16 + m) * LDS_STRIDE + ko];
            acc00 = __builtin_amdgcn_wmma_f32_16x16x4_f32(false, a0, false, b0, (short)0, acc00, false, false);
            acc01 = __builtin_amdgcn_wmma_f32_16x16x4_f32(false, a0, false, b1, (short)0, acc01, false, false);
            acc10 = __builtin_amdgcn_wmma_f32_16x16x4_f32(false, a1, false, b0, (short)0, acc10, false, false);
            acc11 = __builtin_amdgcn_wmma_f32_16x16x4_f32(false, a1, false, b1, (short)0, acc11, false, false);
        }
        __syncthreads();
    }
#endif

    // store: C/D layout — lanes 0-15: M=r, lanes 16-31: M=r+8; col = lane&15
    const int c0 = n0 + wc * 32 + m;
    const int c1 = c0 + 16;
    const int r0 = m0 + wr * 32 + kh * 8;
    #pragma unroll
    for (int r = 0; r < 8; ++r) {
        float o00 = acc00[r], o01 = acc01[r];
        float o10 = acc10[r], o11 = acc11[r];
        if (epi == 1) {
            const float b0v = bias[c0 < N ? c0 : 0];
            const float b1v = bias[c1 < N ? c1 : 0];
            o00 = softplus_f(o00 + b0v);  o01 = softplus_f(o01 + b1v);
            o10 = softplus_f(o10 + b0v);  o11 = softplus_f(o11 + b1v);
        }
        const size_t baseA = (size_t)(r0 + r) * (size_t)N;
        const size_t baseB = (size_t)(r0 + 16 + r) * (size_t)N;
        if (c0 < N) { C[baseA + c0] = o00; C[baseB + c0] = o10; }
        if (c1 < N) { C[baseA + c1] = o01; C[baseB + c1] = o11; }
    }
}

// =====================================================================
// Causal depthwise conv1d (k=4) + bias + SiLU.
// xz: (MROWS, 2*D_INNER); xs half = cols [0, D_INNER). Output xs_act.
// =====================================================================
__global__ __launch_bounds__(256)
void conv_silu_kernel(const float* __restrict__ xz,
                      const float* __restrict__ conv_w,
                      const float* __restrict__ conv_b,
                      float* __restrict__ xs_act)
{
    const int idx = blockIdx.x * 256 + threadIdx.x;   // over MROWS*D_INNER
    const int d = idx % D_INNER;
    const int r = idx / D_INNER;          // r = b*SEQ + l
    const int l = r % SEQ;

    const float4 wv = *(const float4*)(conv_w + d * 4);  // k = 0..3 -> x[l-3..l]
    const size_t rs = (size_t)(2 * D_INNER);
    const float* p = xz + (size_t)r * rs + d;

    float acc = conv_b[d];
    if (l >= 3) acc += wv.x * p[-(ptrdiff_t)(3 * rs)];
    if (l >= 2) acc += wv.y * p[-(ptrdiff_t)(2 * rs)];
    if (l >= 1) acc += wv.z * p[-(ptrdiff_t)(1 * rs)];
    acc += wv.w * p[0];

    xs_act[idx] = silu_f(acc);
}

// =====================================================================
// Selective scan: one lane per (batch, channel). 16 states in registers.
// dt, xs: (MROWS, D_INNER). x_dbl: (MROWS, 96), B = cols 64..79, C = 80..95.
// =====================================================================
__global__ __launch_bounds__(256)
void scan_kernel(const float* __restrict__ dt,
                 const float* __restrict__ xs,
                 const float* __restrict__ xdbl,
                 const float* __restrict__ A_log,
                 float* __restrict__ y)
{
    const int t = blockIdx.x * 256 + threadIdx.x;     // 0 .. MROWS-1 channels
    if (t >= NBATCH * D_INNER) return;
    const int b = t / D_INNER;
    const int d = t % D_INNER;

    float An[D_STATE];
    #pragma unroll
    for (int n = 0; n < D_STATE; ++n)
        An[n] = -__expf(A_log[d * D_STATE + n]);

    float h[D_STATE];
    #pragma unroll
    for (int n = 0; n < D_STATE; ++n) h[n] = 0.f;

    for (int l = 0; l < SEQ; ++l) {
        const size_t row = (size_t)b * SEQ + l;
        const float dtv = dt[row * D_INNER + d];
        const float xv  = xs[row * D_INNER + d];
        const float dtx = dtv * xv;

        const float4* bc = (const float4*)(xdbl + row * XPROJ_N + DT_RANK);
        float Bv[D_STATE], Cv[D_STATE];
        *(float4*)&Bv[0]  = bc[0];  *(float4*)&Bv[4]  = bc[1];
        *(float4*)&Bv[8]  = bc[2];  *(float4*)&Bv[12] = bc[3];
        *(float4*)&Cv[0]  = bc[4];  *(float4*)&Cv[4]  = bc[5];
        *(float4*)&Cv[8]  = bc[6];  *(float4*)&Cv[12] = bc[7];

        float acc = 0.f;
        #pragma unroll
        for (int n = 0; n < D_STATE; ++n) {
            h[n] = h[n] * __expf(dtv * An[n]) + dtx * Bv[n];
            acc += h[n] * Cv[n];
        }
        y[row * D_INNER + d] = acc;
    }
}

// =====================================================================
// y = (y_scan + xs_act * D[d]) * silu(z), z = xz cols [D_INNER, 2*D_INNER)
// In-place on y buffer.
// =====================================================================
__global__ __launch_bounds__(256)
void ymix_kernel(float* __restrict__ y,
                 const float* __restrict__ xs_act,
                 const float* __restrict__ xz,
                 const float* __restrict__ Dvec)
{
    const int idx = blockIdx.x * 256 + threadIdx.x;   // over MROWS*D_INNER
    const int d = idx % D_INNER;
    const int r = idx / D_INNER;
    const float z = xz[(size_t)r * (2 * D_INNER) + D_INNER + d];
    y[idx] = (y[idx] + xs_act[idx] * Dvec[d]) * silu_f(z);
}

// =====================================================================
extern "C" void kernel_launch(void* const* d_in, const int* in_sizes, int n_in,
                              void* d_out, int out_size, void* d_ws, size_t ws_size,
                              hipStream_t stream) {
    (void)in_sizes; (void)n_in; (void)out_size; (void)ws_size;

    const float* x          = (const float*)d_in[0];
    const float* in_proj_w  = (const float*)d_in[1];
    const float* conv_w     = (const float*)d_in[2];
    const float* conv_b     = (const float*)d_in[3];
    const float* x_proj_w   = (const float*)d_in[4];
    const float* dt_proj_w  = (const float*)d_in[5];
    const float* dt_proj_b  = (const float*)d_in[6];
    const float* A_log      = (const float*)d_in[7];
    const float* Dvec       = (const float*)d_in[8];
    const float* out_proj_w = (const float*)d_in[9];
    float* out = (float*)d_out;

    // workspace layout (bytes)
    char* ws = (char*)d_ws;
    float* xz     = (float*)(ws);                    // 4096 x 4096  (64 MB)
    float* xs_act = (float*)(ws +  67108864ull);     // 4096 x 2048  (32 MB)
    float* x_dbl  = (float*)(ws + 100663296ull);     // 4096 x 96    (1.5 MB)
    float* dt     = (float*)(ws + 102236160ull);     // 4096 x 2048  (32 MB)
    float* y      = (float*)(ws + 135790592ull);     // 4096 x 2048  (32 MB)

    const dim3 blk(256);
    const int elemBlocks = (MROWS * D_INNER) / 256;  // 32768
    const int mBlocks = MROWS / 128;                 // 32

    // 1) xz = x @ in_proj_w^T            (M=4096, N=4096, K=1024)
    gemm_wmma_f32<<<dim3(64, mBlocks), blk, 0, stream>>>(
        x, D_MODEL, in_proj_w, nullptr, xz, 2 * D_INNER, D_MODEL, 0);

    // 2) depthwise causal conv + SiLU -> xs_act
    conv_silu_kernel<<<elemBlocks, blk, 0, stream>>>(xz, conv_w, conv_b, xs_act);

    // 3) x_dbl = xs_act @ x_proj_w^T     (N=96, K=2048)
    gemm_wmma_f32<<<dim3(2, mBlocks), blk, 0, stream>>>(
        xs_act, D_INNER, x_proj_w, nullptr, x_dbl, XPROJ_N, D_INNER, 0);

    // 4) dt = softplus(x_dbl[:, :64] @ dt_proj_w^T + b)   (N=2048, K=64, lda=96)
    gemm_wmma_f32<<<dim3(32, mBlocks), blk, 0, stream>>>(
        x_dbl, XPROJ_N, dt_proj_w, dt_proj_b, dt, D_INNER, DT_RANK, 1);

    // 5) selective scan -> y
    scan_kernel<<<dim3((NBATCH * D_INNER) / 256), blk, 0, stream>>>(
        dt, xs_act, x_dbl, A_log, y);

    // 6) y = (y + xs_act*D) * silu(z)
    ymix_kernel<<<elemBlocks, blk, 0, stream>>>(y, xs_act, xz, Dvec);

    // 7) out = y @ out_proj_w^T          (N=1024, K=2048)
    gemm_wmma_f32<<<dim3(16, mBlocks), blk, 0, stream>>>(
        y, D_INNER, out_proj_w, nullptr, out, D_MODEL, D_INNER, 0);
}